// MSAColumnGlobalAttention_38792144617613
// MI455X (gfx1250) — compile-verified
//
#include <hip/hip_runtime.h>
#include <hip/hip_bf16.h>
#include <math.h>

// Problem constants (match reference)
#define N_SEQ 1024
#define N_RES 768
#define C_M   64
#define NUM_HEAD 8
#define HEAD_DIM 8

typedef __attribute__((ext_vector_type(2))) float v2f;
typedef __attribute__((ext_vector_type(8))) float v8f;

__device__ __forceinline__ float wred_sum(float v) {
#pragma unroll
  for (int m = 16; m > 0; m >>= 1) v += __shfl_xor(v, m, 32);
  return v;
}
__device__ __forceinline__ float wred_max(float v) {
#pragma unroll
  for (int m = 16; m > 0; m >>= 1) v = fmaxf(v, __shfl_xor(v, m, 32));
  return v;
}

// ---------------------------------------------------------------------------
// Kernel 1: per column r: LayerNorm, K/V projections, masked q_avg, q = qa*Wq
// grid = N_RES blocks, 256 threads (8 waves). Wave w handles row s0+w.
// ---------------------------------------------------------------------------
__global__ __launch_bounds__(256) void k1_ln_kv_q(
    const float* __restrict__ msa_act, const float* __restrict__ msa_mask,
    const float* __restrict__ ln_scale, const float* __restrict__ ln_bias,
    const float* __restrict__ query_w, const float* __restrict__ key_w,
    const float* __restrict__ value_w,
    float* __restrict__ k_ws, float* __restrict__ v_ws, float* __restrict__ q_ws)
{
  __shared__ float xn[8][68];         // 8 normalized rows, padded vs bank conflicts
  __shared__ float kwl[C_M * HEAD_DIM];
  __shared__ float vwl[C_M * HEAD_DIM];
  __shared__ float qacc[8][C_M];
  __shared__ float qm[8];
  __shared__ float qavg[C_M];
  __shared__ float lsc[C_M], lbs[C_M];

  const int t = threadIdx.x;
  const int w = t >> 5;
  const int lane = t & 31;
  const int r = blockIdx.x;

  if (t < C_M) { lsc[t] = ln_scale[t]; lbs[t] = ln_bias[t]; }
  for (int i = t; i < C_M * HEAD_DIM; i += 256) { kwl[i] = key_w[i]; vwl[i] = value_w[i]; }
  __syncthreads();

  float acc0 = 0.f, acc1 = 0.f, msum = 0.f;

  for (int it = 0; it < N_SEQ / 8; ++it) {
    const int s = it * 8 + w;
    // --- LayerNorm row (s, r): lane owns channels 2*lane, 2*lane+1 (coalesced)
    const float* src = msa_act + ((size_t)s * N_RES + r) * C_M + 2 * lane;
    float x0 = src[0], x1 = src[1];
    float mu  = wred_sum(x0 + x1) * (1.0f / 64.0f);
    float d0 = x0 - mu, d1 = x1 - mu;
    float inv = rsqrtf(wred_sum(d0 * d0 + d1 * d1) * (1.0f / 64.0f) + 1e-5f);
    float y0 = d0 * inv * lsc[2 * lane]     + lbs[2 * lane];
    float y1 = d1 * inv * lsc[2 * lane + 1] + lbs[2 * lane + 1];
    xn[w][2 * lane]     = y0;
    xn[w][2 * lane + 1] = y1;
    const float mv = msa_mask[(size_t)s * N_RES + r];
    acc0 += mv * y0; acc1 += mv * y1; msum += mv;
    __syncthreads();

    // --- K / V projections: 8 rows x 8 outputs x {k,v} = 128 dots
    {
      const int idx = t & 127;
      const int sl = idx >> 3, j = idx & 7;
      const float* wm = (t < 128) ? kwl : vwl;
      float sum = 0.f;
#pragma unroll
      for (int c = 0; c < C_M; ++c) sum += xn[sl][c] * wm[c * HEAD_DIM + j];
      float* dst = (t < 128) ? k_ws : v_ws;
      dst[((size_t)r * N_SEQ + it * 8 + sl) * HEAD_DIM + j] = sum;
    }
    __syncthreads();
  }

  qacc[w][2 * lane] = acc0; qacc[w][2 * lane + 1] = acc1;
  if (lane == 0) qm[w] = msum;
  __syncthreads();
  if (t < C_M) {
    float qs = 0.f, ms = 0.f;
#pragma unroll
    for (int ww = 0; ww < 8; ++ww) { qs += qacc[ww][t]; ms += qm[ww]; }
    qavg[t] = qs / (ms + 1e-10f);
  }
  __syncthreads();
  if (t < C_M) {
    float a = 0.f;
#pragma unroll
    for (int c = 0; c < C_M; ++c) a += qavg[c] * query_w[c * C_M + t];
    q_ws[(size_t)r * C_M + t] = a * 0.35355339059327378f; // HEAD_DIM^-0.5
  }
}

// ---------------------------------------------------------------------------
// Kernel 2: attention: softmax over S per (r, head); wave w == head h.
// grid = N_RES blocks, 256 threads.
// ---------------------------------------------------------------------------
__global__ __launch_bounds__(256) void k2_attn(
    const float* __restrict__ msa_mask,
    const float* __restrict__ k_ws, const float* __restrict__ v_ws,
    const float* __restrict__ q_ws, float* __restrict__ wa_ws)
{
  __shared__ float lg[NUM_HEAD][N_SEQ]; // 32 KB logits
  const int t = threadIdx.x;
  const int h = t >> 5;
  const int lane = t & 31;
  const int r = blockIdx.x;

  float qh[HEAD_DIM];
#pragma unroll
  for (int j = 0; j < HEAD_DIM; ++j) qh[j] = q_ws[(size_t)r * C_M + h * HEAD_DIM + j];

  float lmax = -3.4e38f;
  for (int s = lane; s < N_SEQ; s += 32) {
    const float* kp = k_ws + ((size_t)r * N_SEQ + s) * HEAD_DIM;
    float l = 0.f;
#pragma unroll
    for (int j = 0; j < HEAD_DIM; ++j) l += qh[j] * kp[j];
    l += 1e9f * (msa_mask[(size_t)s * N_RES + r] - 1.0f);
    lg[h][s] = l;
    lmax = fmaxf(lmax, l);
  }
  lmax = wred_max(lmax);

  float es = 0.f;
  float a[HEAD_DIM];
#pragma unroll
  for (int j = 0; j < HEAD_DIM; ++j) a[j] = 0.f;
  for (int s = lane; s < N_SEQ; s += 32) {
    float e = __expf(lg[h][s] - lmax);
    es += e;
    const float* vp = v_ws + ((size_t)r * N_SEQ + s) * HEAD_DIM;
#pragma unroll
    for (int j = 0; j < HEAD_DIM; ++j) a[j] += e * vp[j];
  }
  es = wred_sum(es);
#pragma unroll
  for (int j = 0; j < HEAD_DIM; ++j) a[j] = wred_sum(a[j]);
  if (lane == 0) {
    float inv = 1.0f / es;
#pragma unroll
    for (int j = 0; j < HEAD_DIM; ++j)
      wa_ws[(size_t)r * C_M + h * HEAD_DIM + j] = a[j] * inv;
  }
}

// ---------------------------------------------------------------------------
// Kernel 3: gate GEMM + output GEMM via V_WMMA_F32_16X16X4_F32.
// Weights are pre-swizzled in LDS into B-fragment order so each lane's
// (K, K+1) pair is one aligned ds_load_b64 straight into an even VGPR pair.
// Group g = 2k+hi holds interleaved pairs (W[2g][n], W[2g+1][n]); group
// stride 160 words => hi=0 / hi=1 half-waves hit complementary bank sets.
// grid = (N_SEQ/64, N_RES), 128 threads (4 waves). Wave handles a 16-row tile.
// ---------------------------------------------------------------------------
#define WSWZ_STRIDE 160   // words per fragment group (128 used + 32 pad)

__global__ __launch_bounds__(128) void k3_gate_out(
    const float* __restrict__ msa_act,
    const float* __restrict__ ln_scale, const float* __restrict__ ln_bias,
    const float* __restrict__ gating_w, const float* __restrict__ gating_b,
    const float* __restrict__ output_w, const float* __restrict__ output_b,
    const float* __restrict__ wa_ws, float* __restrict__ out)
{
  __shared__ float wg[32 * WSWZ_STRIDE];  // gating_w, fragment-swizzled (20 KB)
  __shared__ float wo[32 * WSWZ_STRIDE];  // output_w, fragment-swizzled (20 KB)
  __shared__ float xb[4][16][68];         // per-wave 16x64 tile, padded
  __shared__ float lsc[C_M], lbs[C_M], gbl[C_M], obl[C_M], wav[C_M];

  const int t = threadIdx.x;
  const int w = t >> 5;
  const int lane = t & 31;
  const int row = lane & 15;          // fragment row / column index
  const int hi = lane >> 4;           // 0: K=0,1 half; 1: K=2,3 half
  const int r = blockIdx.y;
  const int s0 = (blockIdx.x * 4 + w) * 16;

  // pre-swizzle weights into fragment order
  for (int i = t; i < C_M * C_M; i += 128) {
    const int c = i >> 6, n = i & 63;
    const int idx = (c >> 1) * WSWZ_STRIDE + 2 * n + (c & 1);
    wg[idx] = gating_w[i];
    wo[idx] = output_w[i];
  }
  if (t < C_M) {
    lsc[t] = ln_scale[t]; lbs[t] = ln_bias[t];
    gbl[t] = gating_b[t]; obl[t] = output_b[t];
    wav[t] = wa_ws[(size_t)r * C_M + t];
  }
  __syncthreads();

  // --- LayerNorm 16 rows into xb[w] (coalesced: whole wave loads one row)
  for (int m = 0; m < 16; ++m) {
    const int s = s0 + m;
    const float* src = msa_act + ((size_t)s * N_RES + r) * C_M + 2 * lane;
    float x0 = src[0], x1 = src[1];
    float mu  = wred_sum(x0 + x1) * (1.0f / 64.0f);
    float d0 = x0 - mu, d1 = x1 - mu;
    float inv = rsqrtf(wred_sum(d0 * d0 + d1 * d1) * (1.0f / 64.0f) + 1e-5f);
    xb[w][m][2 * lane]     = d0 * inv * lsc[2 * lane]     + lbs[2 * lane];
    xb[w][m][2 * lane + 1] = d1 * inv * lsc[2 * lane + 1] + lbs[2 * lane + 1];
  }
  __syncthreads();

  const int bcol = 2 * row;           // per-lane column offset in a fragment group

  // --- A fragments: lane<16 holds row, K=4k..4k+1; lane>=16 holds K=4k+2..4k+3
  v2f afr[16];
#pragma unroll
  for (int k = 0; k < 16; ++k)
    afr[k] = *(const v2f*)&xb[w][row][4 * k + 2 * hi];

  // --- gate GEMM: D[16x64] = xn[16x64] @ Wg[64x64]
  v8f acc[4];
#pragma unroll
  for (int tl = 0; tl < 4; ++tl) {
    v8f c = {};
#pragma unroll
    for (int k = 0; k < 16; ++k) {
      v2f b = *(const v2f*)&wg[(2 * k + hi) * WSWZ_STRIDE + 32 * tl + bcol];
      c = __builtin_amdgcn_wmma_f32_16x16x4_f32(false, afr[k], false, b,
                                                (short)0, c, false, false);
    }
    acc[tl] = c;
  }
  __syncthreads();

  // --- sigmoid gate * weighted_avg, write gated back (transposes D->A layout)
#pragma unroll
  for (int tl = 0; tl < 4; ++tl) {
    const int col = tl * 16 + row;
    const float gb = gbl[col];
    const float wv = wav[col];
#pragma unroll
    for (int i = 0; i < 8; ++i) {
      const int m = i + 8 * hi;
      float g = 1.0f / (1.0f + __expf(-(acc[tl][i] + gb)));
      xb[w][m][col] = wv * g;
    }
  }
  __syncthreads();

  // --- output GEMM: D[16x64] = gated[16x64] @ Wo[64x64]
#pragma unroll
  for (int k = 0; k < 16; ++k)
    afr[k] = *(const v2f*)&xb[w][row][4 * k + 2 * hi];
#pragma unroll
  for (int tl = 0; tl < 4; ++tl) {
    v8f c = {};
#pragma unroll
    for (int k = 0; k < 16; ++k) {
      v2f b = *(const v2f*)&wo[(2 * k + hi) * WSWZ_STRIDE + 32 * tl + bcol];
      c = __builtin_amdgcn_wmma_f32_16x16x4_f32(false, afr[k], false, b,
                                                (short)0, c, false, false);
    }
    acc[tl] = c;
  }
  __syncthreads();
#pragma unroll
  for (int tl = 0; tl < 4; ++tl) {
    const int col = tl * 16 + row;
#pragma unroll
    for (int i = 0; i < 8; ++i) xb[w][i + 8 * hi][col] = acc[tl][i];
  }
  __syncthreads();

  // --- coalesced store with output bias: out[s][r][c]
  for (int m = 0; m < 16; ++m) {
    const int s = s0 + m;
    float* dst = out + ((size_t)s * N_RES + r) * C_M + 2 * lane;
    dst[0] = xb[w][m][2 * lane]     + obl[2 * lane];
    dst[1] = xb[w][m][2 * lane + 1] + obl[2 * lane + 1];
  }
}

// ---------------------------------------------------------------------------
extern "C" void kernel_launch(void* const* d_in, const int* in_sizes, int n_in,
                              void* d_out, int out_size, void* d_ws, size_t ws_size,
                              hipStream_t stream) {
  const float* msa_act  = (const float*)d_in[0];
  const float* msa_mask = (const float*)d_in[1];
  const float* ln_scale = (const float*)d_in[2];
  const float* ln_bias  = (const float*)d_in[3];
  const float* query_w  = (const float*)d_in[4];
  const float* key_w    = (const float*)d_in[5];
  const float* value_w  = (const float*)d_in[6];
  const float* gating_w = (const float*)d_in[7];
  const float* gating_b = (const float*)d_in[8];
  const float* output_w = (const float*)d_in[9];
  const float* output_b = (const float*)d_in[10];
  float* out = (float*)d_out;

  float* ws    = (float*)d_ws;
  float* k_ws  = ws;                                        // [R, S, 8]
  float* v_ws  = k_ws + (size_t)N_RES * N_SEQ * HEAD_DIM;   // [R, S, 8]
  float* q_ws  = v_ws + (size_t)N_RES * N_SEQ * HEAD_DIM;   // [R, 64]
  float* wa_ws = q_ws + (size_t)N_RES * C_M;                // [R, 64]

  k1_ln_kv_q<<<N_RES, 256, 0, stream>>>(msa_act, msa_mask, ln_scale, ln_bias,
                                        query_w, key_w, value_w,
                                        k_ws, v_ws, q_ws);
  k2_attn<<<N_RES, 256, 0, stream>>>(msa_mask, k_ws, v_ws, q_ws, wa_ws);
  k3_gate_out<<<dim3(N_SEQ / 64, N_RES), 128, 0, stream>>>(
      msa_act, ln_scale, ln_bias, gating_w, gating_b,
      output_w, output_b, wa_ws, out);
}